// TextureDifference_74534862454843
// MI455X (gfx1250) — compile-verified
//
#include <hip/hip_runtime.h>

// CDNA5 (gfx1250) wave32 WMMA f32 vector types
typedef __attribute__((ext_vector_type(2))) float v2f;   // A/B of 16x16x4 f32: 64 elems / 32 lanes
typedef __attribute__((ext_vector_type(8))) float v8f;   // C/D 16x16 f32: 256 elems / 32 lanes

#define PATCH     5
#define HALO      2
#define TS        16            // output tile (matches WMMA 16x16)
#define RS        (TS + 2*HALO) // 20 = staged region / K dimension
#define IMG_H     512
#define IMG_W     512
#define NBATCH    16
#define CONSTANT_ 1e-5f
#define THRESH_   0.975f
#define VAR_EPS_  1e-9f
#define INV25     0.04f

// reflect padding index (pad width 2, dim 512): -1->1, -2->2, 512->510, 513->509
__device__ __forceinline__ int refl(int i) {
  i = (i < 0) ? -i : i;
  return (i > IMG_W - 1) ? (2 * (IMG_W - 1) - i) : i;
}

__global__ __launch_bounds__(128) void texdiff_wmma_kernel(
    const float* __restrict__ img1, const float* __restrict__ img2,
    float* __restrict__ out)
{
  // per-wave private LDS slices: gray (+sq) staging and horizontal 5-sums
  __shared__ float sG[4][4][RS][RS + 1]; // [wave][chan][row][col] (+1 pad vs bank conflicts)
  __shared__ float sH[4][4][RS][TS];     // horizontal 5-sums: rows 0..19 x cols 0..15

  const int lane = threadIdx.x & 31;
  const int wave = threadIdx.x >> 5;

  // one 16x16 tile per wave; 32x32 tiles per image, 16 images
  const int tileId = blockIdx.x * 4 + wave;
  const int b  = tileId >> 10;        // tile / 1024
  const int t  = tileId & 1023;
  const int oy = (t >> 5) << 4;
  const int ox = (t & 31) << 4;

  const size_t chs   = (size_t)IMG_H * IMG_W;        // channel stride
  const size_t bbase = (size_t)b * 3 * chs;
  const float* __restrict__ p1 = img1 + bbase;
  const float* __restrict__ p2 = img2 + bbase;

  // ---- Step A: grayscale + square over 20x20 reflect-padded region -> LDS ----
  for (int p = lane; p < RS * RS; p += 32) {
    const int r = p / RS, c = p % RS;
    const int y = refl(oy - HALO + r);
    const int x = refl(ox - HALO + c);
    const size_t o = (size_t)y * IMG_W + x;
    const float b1 = p1[o], gc1 = p1[o + chs], r1 = p1[o + 2 * chs];
    const float b2 = p2[o], gc2 = p2[o + chs], r2 = p2[o + 2 * chs];
    const float g1 = 0.144f * b1 + 0.587f * gc1 + 0.299f * r1;
    const float g2 = 0.144f * b2 + 0.587f * gc2 + 0.299f * r2;
    sG[wave][0][r][c] = g1;
    sG[wave][1][r][c] = g1 * g1;
    sG[wave][2][r][c] = g2;
    sG[wave][3][r][c] = g2 * g2;
  }
  __syncthreads();

  // ---- Step B: horizontal 5-sums  H[r][c] = sum_{k=0..4} G[r][c+k] ----
  #pragma unroll
  for (int ch = 0; ch < 4; ++ch) {
    for (int p = lane; p < RS * TS; p += 32) {
      const int r = p / TS, c = p % TS;
      const float* __restrict__ row = &sG[wave][ch][r][0];
      sH[wave][ch][r][c] = row[c] + row[c + 1] + row[c + 2] + row[c + 3] + row[c + 4];
    }
  }
  __syncthreads();

  // ---- Step C: vertical banded mean via V_WMMA_F32_16X16X4_F32 ----
  // D(16x16) = M(16x20) x H(20x16), M[i][r] = (i <= r <= i+4) ? 1/25 : 0
  // A layout (16x4 f32): lane = row M (lane&15); VGPR0 = K0 (lanes 0-15) / K2 (16-31),
  //                      VGPR1 = K1 / K3.  B mirrors with lane = col N.
  const int half = lane >> 4;   // 0: K even base, 1: K even base + 2
  const int mn   = lane & 15;   // A row / B col / D col

  v8f acc[4];
  #pragma unroll
  for (int ch = 0; ch < 4; ++ch)
    #pragma unroll
    for (int j = 0; j < 8; ++j) acc[ch][j] = 0.0f;

  #pragma unroll
  for (int kc = 0; kc < 5; ++kc) {
    const int k0 = kc * 4 + 2 * half;     // K index held in .x (then .y = k0+1)
    v2f a;
    a.x = (k0     >= mn && k0     <= mn + 4) ? INV25 : 0.0f;
    a.y = (k0 + 1 >= mn && k0 + 1 <= mn + 4) ? INV25 : 0.0f;
    #pragma unroll
    for (int ch = 0; ch < 4; ++ch) {
      v2f bv;
      bv.x = sH[wave][ch][k0][mn];
      bv.y = sH[wave][ch][k0 + 1][mn];
      // (neg_a, A, neg_b, B, c_mod, C, reuse_a, reuse_b)
      acc[ch] = __builtin_amdgcn_wmma_f32_16x16x4_f32(
          false, a, false, bv, (short)0, acc[ch], false, false);
    }
  }

  // ---- Step D: per-lane epilogue over 8 D values (M = v + 8*half, N = mn) ----
  float* __restrict__ po = out + (size_t)b * chs;
  #pragma unroll
  for (int v = 0; v < 8; ++v) {
    const int m = v + 8 * half;
    const float m1  = acc[0][v];
    const float m1s = acc[1][v];
    const float m2  = acc[2][v];
    const float m2s = acc[3][v];
    const float var1 = fmaxf(m1s - m1 * m1, 0.0f);
    const float var2 = fmaxf(m2s - m2 * m2, 0.0f);
    const float s1 = sqrtf(var1 + VAR_EPS_);
    const float s2 = sqrtf(var2 + VAR_EPS_);
    const float sim = (2.0f * s1 * s2) / (s1 * s1 + s2 * s2 + CONSTANT_);
    po[(size_t)(oy + m) * IMG_W + (ox + mn)] = (sim > THRESH_) ? 1.0f : 0.0f;
  }
}

extern "C" void kernel_launch(void* const* d_in, const int* in_sizes, int n_in,
                              void* d_out, int out_size, void* d_ws, size_t ws_size,
                              hipStream_t stream) {
  const float* img1 = (const float*)d_in[0];
  const float* img2 = (const float*)d_in[1];
  float* out = (float*)d_out;
  // 16 images * 32*32 tiles = 16384 tiles, 4 tiles (waves) per 128-thread block
  const int blocks = (NBATCH * (IMG_H / TS) * (IMG_W / TS)) / 4;
  hipLaunchKernelGGL(texdiff_wmma_kernel, dim3(blocks), dim3(128), 0, stream,
                     img1, img2, out);
}